// ClasswiseECELoss_1125281432121
// MI455X (gfx1250) — compile-verified
//
#include <hip/hip_runtime.h>
#include <hip/hip_bf16.h>
#include <math.h>

typedef __attribute__((ext_vector_type(2))) float v2f;
typedef __attribute__((ext_vector_type(8))) float v8f;

#define NBINS 10
#define C_CLASSES 1000
#define SEGS (C_CLASSES * NBINS)          /* 10000 global segments (c*10+bin) */
#define LDS_STRIDE 11                      /* pad 10->11 so lane stride 11 is coprime with 64 banks */
#define LDS_SEGS (C_CLASSES * LDS_STRIDE)  /* 11000 */

// ---------------------------------------------------------------------------
// Kernel 0: zero the global accumulators in workspace (cnt | conf | acc)
// ---------------------------------------------------------------------------
__global__ void sce_zero_kernel(float* __restrict__ ws, int n) {
  int i = blockIdx.x * blockDim.x + threadIdx.x;
  if (i < n) ws[i] = 0.0f;
}

// ---------------------------------------------------------------------------
// Kernel 1: streaming softmax + (class,bin) histogram.
// One wave per row. Lane l owns classes {l, l+32, ..., l+992} -> 32 regs.
// LDS histogram padded to 11 floats/class so all 32 lanes of a wave hit
// distinct LDS banks on ds_add_f32 (stride 11, gcd(11,64)=1).
// ---------------------------------------------------------------------------
__global__ __launch_bounds__(1024) void sce_hist_kernel(
    const float* __restrict__ logits, const int* __restrict__ labels,
    float* __restrict__ g_cnt, float* __restrict__ g_conf,
    float* __restrict__ g_acc, int N) {
  __shared__ float sh_cnt[LDS_SEGS];
  __shared__ float sh_conf[LDS_SEGS];
  for (int i = threadIdx.x; i < LDS_SEGS; i += blockDim.x) {
    sh_cnt[i] = 0.0f;
    sh_conf[i] = 0.0f;
  }
  __syncthreads();

  const int lane = threadIdx.x & 31;
  const int wavesPerBlock = blockDim.x >> 5;
  const int gwave = blockIdx.x * wavesPerBlock + (threadIdx.x >> 5);
  const int nwaves = gridDim.x * wavesPerBlock;
  const int NCHUNK = (C_CLASSES + 31) / 32;  /* 32; last chunk partial (8 lanes) */

  for (int row = gwave; row < N; row += nwaves) {
    const float* rp = logits + (size_t)row * C_CLASSES;
    float val[32];
    float m = -__builtin_inff();
#pragma unroll
    for (int j = 0; j < 32; ++j) {
      int c = j * 32 + lane;
      if (c < C_CLASSES) {
        val[j] = rp[c];
        m = fmaxf(m, val[j]);
      } else {
        val[j] = -__builtin_inff();
      }
    }
    // wave32 max reduction
#pragma unroll
    for (int off = 16; off > 0; off >>= 1) m = fmaxf(m, __shfl_xor(m, off, 32));

    float s = 0.0f;
#pragma unroll
    for (int j = 0; j < 32; ++j) {
      int c = j * 32 + lane;
      if (c < C_CLASSES) {
        val[j] = __expf(val[j] - m);
        s += val[j];
      }
    }
    // wave32 sum reduction
#pragma unroll
    for (int off = 16; off > 0; off >>= 1) s += __shfl_xor(s, off, 32);
    const float inv = 1.0f / s;
    const int   L   = labels[row];

#pragma unroll
    for (int j = 0; j < 32; ++j) {
      int c = j * 32 + lane;
      if (c < C_CLASSES) {
        float p = val[j] * inv;
        if (p > 0.0f) {  /* exactly-0 confidences fall in no bin */
          int bin = (int)ceilf(p * (float)NBINS) - 1;
          bin = min(max(bin, 0), NBINS - 1);
          int lidx = c * LDS_STRIDE + bin;  /* conflict-free across lanes */
          atomicAdd(&sh_cnt[lidx], 1.0f);
          atomicAdd(&sh_conf[lidx], p);
          if (c == L) atomicAdd(&g_acc[c * NBINS + bin], 1.0f);
        }
      }
    }
  }

  __syncthreads();
  // Flush block-private histogram; skip empty bins (almost all of them:
  // with ~1/C probs nearly everything lands in bin 0 of each class).
  for (int seg = threadIdx.x; seg < SEGS; seg += blockDim.x) {
    int c = seg / NBINS, b = seg - c * NBINS;
    float cv = sh_cnt[c * LDS_STRIDE + b];
    if (cv != 0.0f) {
      atomicAdd(&g_cnt[seg], cv);
      atomicAdd(&g_conf[seg], sh_conf[c * LDS_STRIDE + b]);
    }
  }
}

// ---------------------------------------------------------------------------
// Kernel 2: finalize. per_bin = |conf/cnt - acc/cnt| * (cnt/N), total/C.
// 256 threads -> 256 partials -> wave0 sums them with v_wmma_f32_16x16x4_f32
// (A = all-ones => every row of D holds column sums; total sum is layout-
// independent), then a 32-lane shuffle sum. lanes 0-15 hold row0, 16-31 row8
// (identical), so sum over 32 lanes and scale by 0.5.
// ---------------------------------------------------------------------------
__global__ __launch_bounds__(256) void sce_finalize_kernel(
    const float* __restrict__ g_cnt, const float* __restrict__ g_conf,
    const float* __restrict__ g_acc, float* __restrict__ out, float invN) {
  __shared__ float partial[256];
  float acc = 0.0f;
  for (int i = threadIdx.x; i < SEGS; i += 256) {
    float cnt = g_cnt[i];
    if (cnt > 0.0f) {
      float rc  = 1.0f / cnt;
      float gap = fabsf(g_conf[i] * rc - g_acc[i] * rc);
      acc += gap * (cnt * invN);
    }
  }
  partial[threadIdx.x] = acc;
  __syncthreads();

  if (threadIdx.x < 32) {
    v2f a; a.x = 1.0f; a.y = 1.0f;   /* A(16x4) = all ones */
    v8f cm = {};                      /* C/D accumulator    */
#pragma unroll
    for (int chunk = 0; chunk < 4; ++chunk) {
      v2f b;
      b.x = partial[chunk * 64 + threadIdx.x];
      b.y = partial[chunk * 64 + 32 + threadIdx.x];
      // 8 args: (neg_a, A, neg_b, B, c_mod, C, reuse_a, reuse_b)
      cm = __builtin_amdgcn_wmma_f32_16x16x4_f32(
          false, a, false, b, (short)0, cm, false, false);
    }
    float r = cm[0];
#pragma unroll
    for (int off = 16; off > 0; off >>= 1) r += __shfl_xor(r, off, 32);
    if (threadIdx.x == 0) out[0] = (r * 0.5f) / (float)C_CLASSES;
  }
}

// ---------------------------------------------------------------------------
extern "C" void kernel_launch(void* const* d_in, const int* in_sizes, int n_in,
                              void* d_out, int out_size, void* d_ws, size_t ws_size,
                              hipStream_t stream) {
  const float* logits = (const float*)d_in[0];
  const int*   labels = (const int*)d_in[1];  /* labels (int32 per harness) */
  const int N = in_sizes[1];                  /* 131072 rows */

  float* ws     = (float*)d_ws;
  float* g_cnt  = ws;                 /* [SEGS] */
  float* g_conf = ws + SEGS;          /* [SEGS] */
  float* g_acc  = ws + 2 * SEGS;      /* [SEGS] */

  const int zn = 3 * SEGS;
  sce_zero_kernel<<<(zn + 255) / 256, 256, 0, stream>>>(ws, zn);

  // 512 blocks x 1024 threads = 16384 waves; 8 rows per wave (grid-stride).
  sce_hist_kernel<<<512, 1024, 0, stream>>>(logits, labels, g_cnt, g_conf,
                                            g_acc, N);

  sce_finalize_kernel<<<1, 256, 0, stream>>>(g_cnt, g_conf, g_acc,
                                             (float*)d_out, 1.0f / (float)N);
}